// MOTNet_12498354831484
// MI455X (gfx1250) — compile-verified
//
#include <hip/hip_runtime.h>
#include <math.h>

typedef __attribute__((ext_vector_type(2))) float v2f;
typedef __attribute__((ext_vector_type(8))) float v8f;

#define Gn    128
#define NPER  512
#define ETOT  (128 * 4096)   // 524288
#define FIN   64
#define Hh    128
#define K1n   410
#define K2n   328
#define N1    (Gn * NPER)    // 65536
#define N2    (Gn * K1n)     // 52480
#define N3    (Gn * K2n)     // 41984

// ------- WMMA fp32 GEMM: D = A[MxK] @ B[KxN] (+Cin) (+bias) (relu), N == NT*16 -------
// One 16xN row-strip per wave (wave32): NT accumulators share one A fragment per
// k-step. B fragments for the whole k-step are batch-loaded first (one clause,
// one wait), then NT independent WMMAs run back-to-back (no D->A/B hazards).
// Cin is added in the epilogue (additive identity) to keep loop liveness at
// 64 accum + 16 B + 2 A VGPRs and avoid prologue spills.
// A-frag per ISA 16x4 f32 layout: VGPR0 = K{0|2} per lane-half, VGPR1 = K{1|3};
// B mirrors (row striped across lanes).
template <int NT>
__global__ void gemm16_wmma_f32_nt(const float* __restrict__ A,
                                   const float* __restrict__ B,
                                   const float* __restrict__ Cin,   // nullable
                                   const float* __restrict__ bias,  // nullable
                                   float* __restrict__ D,
                                   int M, int K, int N, int doRelu)
{
    const int lane = threadIdx.x & 31;
    const int wave = threadIdx.x >> 5;
    const int rowTiles = M >> 4;
    const int tile = blockIdx.x * (blockDim.x >> 5) + wave;
    if (tile >= rowTiles) return;                 // wave-uniform: EXEC all-1 for WMMA

    const int m0 = tile << 4;
    const int half = lane >> 4;                   // 0: lanes 0-15, 1: lanes 16-31
    const int r    = lane & 15;

    v8f acc[NT];
#pragma unroll
    for (int t = 0; t < NT; ++t)
        acc[t] = (v8f){0.f, 0.f, 0.f, 0.f, 0.f, 0.f, 0.f, 0.f};

    const float* arow = A + (size_t)(m0 + r) * K;
#pragma unroll 2
    for (int kk = 0; kk < K; kk += 4) {
        v2f a;
        a.x = arow[kk + 2 * half + 0];
        a.y = arow[kk + 2 * half + 1];
        const float* brow0 = B + (size_t)(kk + 2 * half + 0) * N + r;
        const float* brow1 = B + (size_t)(kk + 2 * half + 1) * N + r;
        v2f bfrag[NT];
#pragma unroll
        for (int t = 0; t < NT; ++t) {            // batch loads -> single clause/wait
            bfrag[t].x = brow0[t * 16];
            bfrag[t].y = brow1[t * 16];
        }
#pragma unroll
        for (int t = 0; t < NT; ++t)              // NT back-to-back independent WMMAs
            acc[t] = __builtin_amdgcn_wmma_f32_16x16x4_f32(false, a, false, bfrag[t],
                                                           (short)0, acc[t], false, false);
    }

#pragma unroll
    for (int t = 0; t < NT; ++t) {
        const float bv = bias ? bias[t * 16 + r] : 0.f;
#pragma unroll
        for (int v = 0; v < 8; ++v) {
            const size_t idx = (size_t)(m0 + v + 8 * half) * N + t * 16 + r;
            float val = acc[t][v] + bv;
            if (Cin) val += Cin[idx];
            if (doRelu) val = fmaxf(val, 0.f);
            D[idx] = val;
        }
    }
}

// ---------------- helpers ----------------
__global__ void zero_kernel(float* __restrict__ buf, size_t n)
{
    size_t i = (size_t)blockIdx.x * blockDim.x + threadIdx.x;
    if (i < n) buf[i] = 0.f;
}

// agg[dst] += xr[src] for every live edge; -1 sentinels skip (masked edges)
__global__ void edge_scatter_kernel(const float* __restrict__ xr,
                                    const int* __restrict__ src,
                                    const int* __restrict__ dst,
                                    float* __restrict__ agg)
{
    const int e = blockIdx.x;
    const int f = threadIdx.x;                    // 128 = H
    const int s = src[e], d = dst[e];
    if (s >= 0 && d >= 0)
        atomicAdd(&agg[(size_t)d * Hh + f], xr[(size_t)s * Hh + f]);
}

__global__ void pnorm_kernel(const float* __restrict__ p, float* __restrict__ out)
{
    __shared__ float red[Hh];
    const int t = threadIdx.x;
    const float v = p[t];
    red[t] = v * v;
    __syncthreads();
    for (int s = Hh / 2; s > 0; s >>= 1) {
        if (t < s) red[t] += red[t + s];
        __syncthreads();
    }
    if (t == 0) out[0] = rsqrtf(red[0]);
}

__global__ void score_kernel(const float* __restrict__ h, const float* __restrict__ p,
                             const float* __restrict__ invnorm,
                             float* __restrict__ score, int nNodes)
{
    const int i = blockIdx.x * blockDim.x + threadIdx.x;
    if (i >= nNodes) return;
    const float* row = h + (size_t)i * Hh;
    float acc = 0.f;
    for (int f = 0; f < Hh; ++f) acc += row[f] * p[f];
    score[i] = tanhf(acc * invnorm[0]);
}

// per-graph descending bitonic sort of 512 (score,idx) pairs in LDS; keep top-k
__global__ void topk_kernel(const float* __restrict__ score,
                            float* __restrict__ vals,
                            int* __restrict__ oldIdx,
                            int* __restrict__ mapping,   // nullable
                            int nPer, int k)
{
    __shared__ float s[512];
    __shared__ int   id[512];
    const int g = blockIdx.x;
    const int tid = threadIdx.x;                  // blockDim = 256
    for (int t = tid; t < 512; t += 256) {
        s[t]  = (t < nPer) ? score[(size_t)g * nPer + t] : -__builtin_inff();
        id[t] = t;
        if (mapping && t < nPer) mapping[(size_t)g * nPer + t] = -1;
    }
    __syncthreads();
    for (int kk = 2; kk <= 512; kk <<= 1) {
        for (int j = kk >> 1; j > 0; j >>= 1) {
            for (int t = tid; t < 512; t += 256) {
                const int ixj = t ^ j;
                if (ixj > t) {
                    const bool desc = ((t & kk) == 0);
                    const float a = s[t], b = s[ixj];
                    if (desc ? (a < b) : (a > b)) {
                        s[t] = b; s[ixj] = a;
                        const int tmp = id[t]; id[t] = id[ixj]; id[ixj] = tmp;
                    }
                }
            }
            __syncthreads();
        }
    }
    for (int t = tid; t < k; t += 256) {
        const int oldGlobal = g * nPer + id[t];
        oldIdx[(size_t)g * k + t] = oldGlobal;
        vals[(size_t)g * k + t]   = s[t];
        if (mapping) mapping[oldGlobal] = g * k + t;
    }
}

__global__ void gather_scale_kernel(const float* __restrict__ h,
                                    const int* __restrict__ oldIdx,
                                    const float* __restrict__ vals,
                                    float* __restrict__ out)
{
    const int i = blockIdx.x;                     // new node id
    const int f = threadIdx.x;                    // 128
    out[(size_t)i * Hh + f] = h[(size_t)oldIdx[i] * Hh + f] * vals[i];
}

__global__ void remap_edges_kernel(const int* __restrict__ src, const int* __restrict__ dst,
                                   const int* __restrict__ mapping,
                                   int* __restrict__ ns, int* __restrict__ nd, int E)
{
    const int e = blockIdx.x * blockDim.x + threadIdx.x;
    if (e >= E) return;
    ns[e] = mapping[src[e]];
    nd[e] = mapping[dst[e]];
}

__global__ void readout_kernel(const float* __restrict__ hp, float* __restrict__ out, int k)
{
    const int g = blockIdx.x;
    const int f = threadIdx.x;                    // 128
    const float* base = hp + (size_t)g * k * Hh + f;
    float m = -__builtin_inff(), sum = 0.f;
    for (int j = 0; j < k; ++j) {
        const float v = base[(size_t)j * Hh];
        m = fmaxf(m, v);
        sum += v;
    }
    out[(size_t)g * (2 * Hh) + f]      = m;
    out[(size_t)g * (2 * Hh) + Hh + f] = sum / (float)k;
}

__global__ void add_kernel(const float* __restrict__ a, const float* __restrict__ b,
                           float* __restrict__ c, int n)
{
    const int i = blockIdx.x * blockDim.x + threadIdx.x;
    if (i < n) c[i] = a[i] + b[i];
}

__global__ void final_kernel(const float* __restrict__ z2, const float* __restrict__ W,
                             const float* __restrict__ b, float* __restrict__ out)
{
    const int g = threadIdx.x;                    // 128 graphs
    float acc = b[0];
    for (int j = 0; j < 64; ++j) acc += z2[(size_t)g * 64 + j] * W[j];
    out[g] = 1.f / (1.f + expf(-acc));
}

// ---------------- launch ----------------
extern "C" void kernel_launch(void* const* d_in, const int* in_sizes, int n_in,
                              void* d_out, int out_size, void* d_ws, size_t ws_size,
                              hipStream_t stream)
{
    const float* x       = (const float*)d_in[0];
    const int*   esrc    = (const int*)  d_in[1];
    const int*   edst    = (const int*)  d_in[2];
    const float* W1_root = (const float*)d_in[3];
    const float* W1_rel  = (const float*)d_in[4];
    const float* b1      = (const float*)d_in[5];
    const float* p1      = (const float*)d_in[6];
    const float* W2_root = (const float*)d_in[7];
    const float* W2_rel  = (const float*)d_in[8];
    const float* b2      = (const float*)d_in[9];
    const float* p2      = (const float*)d_in[10];
    const float* lin1_W  = (const float*)d_in[11];
    const float* lin1_b  = (const float*)d_in[12];
    const float* lin2_W  = (const float*)d_in[13];
    const float* lin2_b  = (const float*)d_in[14];
    const float* lin3_W  = (const float*)d_in[15];
    const float* lin3_b  = (const float*)d_in[16];
    float* out = (float*)d_out;

    // workspace carve-up (fp32 first, then ints)
    float* ws  = (float*)d_ws;
    size_t o = 0;
    float* xr   = ws + o; o += (size_t)N1 * Hh;   // rel-GEMM out; reused as xr2 / h2p
    float* agg  = ws + o; o += (size_t)N1 * Hh;   // segment-sum accumulator
    float* hbuf = ws + o; o += (size_t)N1 * Hh;   // h1, then h2
    float* hp   = ws + o; o += (size_t)N2 * Hh;   // pooled h1
    float* sc   = ws + o; o += N1;                // scores
    float* vals = ws + o; o += N1;                // topk values
    float* x1   = ws + o; o += Gn * 2 * Hh;
    float* x2   = ws + o; o += Gn * 2 * Hh;
    float* zin  = ws + o; o += Gn * 2 * Hh;
    float* z1   = ws + o; o += Gn * Hh;
    float* z2   = ws + o; o += Gn * 64;
    float* pn   = ws + o; o += 4;
    int* oldi    = (int*)(ws + o);
    int* mapping = oldi + N1;
    int* ns      = mapping + N1;
    int* nd      = ns + ETOT;
    (void)in_sizes; (void)n_in; (void)out_size; (void)ws_size;

    auto gemm = [&](const float* A, const float* B, const float* Cin, const float* bias,
                    float* D, int M, int K, int N, int relu) {
        const int rowTiles = M / 16;
        const int blocks   = (rowTiles + 7) / 8;  // 8 waves (wave32) per block
        if (N == 128)
            gemm16_wmma_f32_nt<8><<<blocks, 256, 0, stream>>>(A, B, Cin, bias, D, M, K, N, relu);
        else
            gemm16_wmma_f32_nt<4><<<blocks, 256, 0, stream>>>(A, B, Cin, bias, D, M, K, N, relu);
    };

    // ---- GraphConv 1: h1 = relu(x@W1_root + scatter(x@W1_rel) + b1) ----
    zero_kernel<<<((size_t)N1 * Hh + 255) / 256, 256, 0, stream>>>(agg, (size_t)N1 * Hh);
    gemm(x, W1_rel, nullptr, nullptr, xr, N1, FIN, Hh, 0);
    edge_scatter_kernel<<<ETOT, Hh, 0, stream>>>(xr, esrc, edst, agg);
    gemm(x, W1_root, agg, b1, hbuf, N1, FIN, Hh, 1);

    // ---- TopK pool 1 ----
    pnorm_kernel<<<1, Hh, 0, stream>>>(p1, pn);
    score_kernel<<<(N1 + 255) / 256, 256, 0, stream>>>(hbuf, p1, pn, sc, N1);
    topk_kernel<<<Gn, 256, 0, stream>>>(sc, vals, oldi, mapping, NPER, K1n);
    gather_scale_kernel<<<N2, Hh, 0, stream>>>(hbuf, oldi, vals, hp);
    readout_kernel<<<Gn, Hh, 0, stream>>>(hp, x1, K1n);
    remap_edges_kernel<<<(ETOT + 255) / 256, 256, 0, stream>>>(esrc, edst, mapping, ns, nd, ETOT);

    // ---- GraphConv 2: h2 = relu(hp@W2_root + scatter(hp@W2_rel, masked) + b2) ----
    zero_kernel<<<((size_t)N2 * Hh + 255) / 256, 256, 0, stream>>>(agg, (size_t)N2 * Hh);
    gemm(hp, W2_rel, nullptr, nullptr, xr, N2, Hh, Hh, 0);
    edge_scatter_kernel<<<ETOT, Hh, 0, stream>>>(xr, ns, nd, agg);
    gemm(hp, W2_root, agg, b2, hbuf, N2, Hh, Hh, 1);

    // ---- TopK pool 2 (pad 410 -> 512 with -inf) ----
    pnorm_kernel<<<1, Hh, 0, stream>>>(p2, pn);
    score_kernel<<<(N2 + 255) / 256, 256, 0, stream>>>(hbuf, p2, pn, sc, N2);
    topk_kernel<<<Gn, 256, 0, stream>>>(sc, vals, oldi, nullptr, K1n, K2n);
    gather_scale_kernel<<<N3, Hh, 0, stream>>>(hbuf, oldi, vals, xr);   // xr reused as h2p
    readout_kernel<<<Gn, Hh, 0, stream>>>(xr, x2, K2n);

    // ---- MLP head ----
    add_kernel<<<(Gn * 2 * Hh + 255) / 256, 256, 0, stream>>>(x1, x2, zin, Gn * 2 * Hh);
    gemm(zin, lin1_W, nullptr, lin1_b, z1, Gn, 2 * Hh, Hh, 1);
    gemm(z1,  lin2_W, nullptr, lin2_b, z2, Gn, Hh, 64, 1);
    final_kernel<<<1, Gn, 0, stream>>>(z2, lin3_W, lin3_b, out);
}